// CRF_11871289606632
// MI455X (gfx1250) — compile-verified
//
#include <hip/hip_runtime.h>
#include <math.h>

typedef __attribute__((ext_vector_type(2))) float v2f;
typedef __attribute__((ext_vector_type(8))) float v8f;
typedef __attribute__((ext_vector_type(4))) int   v4i;

#define B_   64
#define T_   512
#define L_   48
#define MB   16          // batches per workgroup (WMMA M dim)
#define NG   (B_/MB)     // 4 workgroups
#define START_ 46
#define PAD_   45
#define END_   47

#define HAS_ASYNC_LDS (__has_builtin(__builtin_amdgcn_global_load_async_to_lds_b128) && \
                       __has_builtin(__builtin_amdgcn_s_wait_asynccnt))

// One wave32 per block; each block owns MB=16 CRF chains.
// Per time step: D[16x48] = exp(a - rowmax) @ exp(trans)  via 36x V_WMMA_F32_16X16X4_F32,
// then fs = rowmax + log(D). Scores for step t+1 are double-buffered into LDS with
// GLOBAL_LOAD_ASYNC_TO_LDS_B128 (ASYNCcnt) while step t computes.
__global__ __launch_bounds__(32) void crf_fwd_kernel(
    const float* __restrict__ scores,   // [B,T,L]
    const int*   __restrict__ gold,     // [B,T]
    const int*   __restrict__ mask,     // [B,T]
    const float* __restrict__ trans,    // [L,L]
    float* __restrict__ partial)        // [NG]
{
    __shared__ float expT[L_][L_];                  // exp(transitions), constant across steps
    __shared__ float aE[MB][L_ + 1];                // staging: a, then exp(a - m) (+1 bank pad)
    __shared__ float fs[MB][L_];                    // forward scores
    __shared__ alignas(16) float srow[2][MB][L_];   // double-buffered score tiles
    __shared__ float rowm[MB];
    __shared__ int   mk[MB];
    __shared__ float red[MB];

    const int lane = threadIdx.x;       // 0..31
    const int g    = blockIdx.x;        // batch group
    const int half = lane >> 4;         // 0/1 : WMMA K/M-half selector
    const int lm   = lane & 15;

    // exp(transitions) (exp(-10000) underflows to exactly 0), init fs = trans[START,:]
    for (int idx = lane; idx < L_ * L_; idx += 32)
        expT[idx / L_][idx % L_] = expf(trans[idx]);
    for (int idx = lane; idx < MB * L_; idx += 32)
        fs[idx / L_][idx % L_] = trans[START_ * L_ + (idx % L_)];

#if HAS_ASYNC_LDS
    // Preload the t=0 score tile into buffer 0 (6x B128 per wave, 16B-aligned).
    {
        #pragma unroll
        for (int it = 0; it < 6; ++it) {
            int q = it * 32 + lane;           // quad index 0..191 over the 16x48 tile
            int b = q / 12;                   // 12 quads per 48-float row
            int i = (q % 12) * 4;
            __builtin_amdgcn_global_load_async_to_lds_b128(
                (v4i*)(scores + ((size_t)(g * MB + b) * T_ + 0) * L_ + i),
                (v4i*)&srow[0][b][i], 0, 0);
        }
    }
#endif
    __syncthreads();

    // Hoist all B-fragments of exp(trans) into registers; reused for all 512 steps.
    // B(4x16) layout: lane l -> (K = 2*half + {0,1}, N = l%16)
    v2f bf[12][3];
    #pragma unroll
    for (int kk = 0; kk < 12; ++kk) {
        #pragma unroll
        for (int tt = 0; tt < 3; ++tt) {
            bf[kk][tt].x = expT[4 * kk + 2 * half    ][16 * tt + lm];
            bf[kk][tt].y = expT[4 * kk + 2 * half + 1][16 * tt + lm];
        }
    }

    int cur = 0;
    for (int t = 0; t < T_; ++t) {
#if HAS_ASYNC_LDS
        // Wait for this step's tile, then stage a = fs + scores from LDS.
        __builtin_amdgcn_s_wait_asynccnt(0);
        __syncthreads();
        for (int idx = lane; idx < MB * L_; idx += 32) {
            int b = idx / L_, i = idx % L_;
            aE[b][i] = fs[b][i] + srow[cur][b][i];
        }
        // Kick off the async fill of the other buffer for step t+1.
        if (t + 1 < T_) {
            #pragma unroll
            for (int it = 0; it < 6; ++it) {
                int q = it * 32 + lane;
                int b = q / 12;
                int i = (q % 12) * 4;
                __builtin_amdgcn_global_load_async_to_lds_b128(
                    (v4i*)(scores + ((size_t)(g * MB + b) * T_ + (t + 1)) * L_ + i),
                    (v4i*)&srow[cur ^ 1][b][i], 0, 0);
            }
        }
#else
        // Fallback: synchronous staging with a software prefetch of the next row.
        const float* srow_g = scores + ((size_t)(g * MB) * T_ + t) * L_;
        for (int idx = lane; idx < MB * L_; idx += 32) {
            int b = idx / L_, i = idx % L_;
            aE[b][i] = fs[b][i] + srow_g[(size_t)b * T_ * L_ + i];
        }
        if (t + 1 < T_) __builtin_prefetch(srow_g + L_, 0, 3);   // global_prefetch_b8
#endif
        if (lane < MB) mk[lane] = mask[(size_t)(g * MB + lane) * T_ + t];
        __syncthreads();

        // per-chain max (global max over i is a valid logsumexp shift)
        if (lane < MB) {
            float m = -INFINITY;
            #pragma unroll 4
            for (int i = 0; i < L_; ++i) m = fmaxf(m, aE[lane][i]);
            rowm[lane] = m;
        }
        __syncthreads();

        // E = exp(a - m)  in (0, 1]
        for (int idx = lane; idx < MB * L_; idx += 32) {
            int b = idx / L_, i = idx % L_;
            aE[b][i] = expf(aE[b][i] - rowm[b]);
        }
        __syncthreads();

        // D[16x48] = E[16x48] @ expT[48x48]   (f32 WMMA, K in chunks of 4)
        v8f acc0 = {}, acc1 = {}, acc2 = {};
        #pragma unroll
        for (int kk = 0; kk < 12; ++kk) {
            v2f a;  // A(16x4): lane l -> (M = l%16, K = 2*half + {0,1})
            a.x = aE[lm][4 * kk + 2 * half];
            a.y = aE[lm][4 * kk + 2 * half + 1];
            acc0 = __builtin_amdgcn_wmma_f32_16x16x4_f32(false, a, false, bf[kk][0],
                                                         (short)0, acc0, false, false);
            acc1 = __builtin_amdgcn_wmma_f32_16x16x4_f32(false, a, false, bf[kk][1],
                                                         (short)0, acc1, false, false);
            acc2 = __builtin_amdgcn_wmma_f32_16x16x4_f32(false, a, false, bf[kk][2],
                                                         (short)0, acc2, false, false);
        }
        __syncthreads();

        // fs[b][j] = m_b + log(D[b][j])  where mask==1 (else keep old fs)
        // C/D layout: VGPR r, lane l -> (M = r + 8*half, N = l%16)
        #pragma unroll
        for (int r = 0; r < 8; ++r) {
            int b = r + 8 * half;
            if (mk[b]) {
                fs[b][ 0 + lm] = rowm[b] + logf(acc0[r]);
                fs[b][16 + lm] = rowm[b] + logf(acc1[r]);
                fs[b][32 + lm] = rowm[b] + logf(acc2[r]);
            }
        }
        __syncthreads();
        cur ^= 1;
    }

    // Gold-path energy (reference flat-index semantics: row 0 of crf):
    // tg_b = trans[0,START] + sum_t mask[b,t] * (scores[b,t,0] + trans[0, gold[b,t]])
    if (lane < MB) {
        int bb = g * MB + lane;
        float tg = trans[0 * L_ + START_];
        for (int t = 0; t < T_; ++t) {
            int   gd = gold[(size_t)bb * T_ + t];
            int   mm = mask[(size_t)bb * T_ + t];
            float term = scores[((size_t)bb * T_ + t) * L_ + 0] + trans[gd];
            tg += mm ? term : 0.0f;
        }
        red[lane] = fs[lane][END_] - tg;
    }
    __syncthreads();
    if (lane == 0) {
        float s = 0.0f;
        for (int i = 0; i < MB; ++i) s += red[i];
        partial[g] = s;
    }
}

__global__ void crf_final_kernel(const float* __restrict__ partial,
                                 float* __restrict__ out)
{
    if (threadIdx.x == 0 && blockIdx.x == 0) {
        float s = 0.0f;
        for (int i = 0; i < NG; ++i) s += partial[i];   // fixed order -> deterministic
        out[0] = s / (float)B_;
    }
}

extern "C" void kernel_launch(void* const* d_in, const int* in_sizes, int n_in,
                              void* d_out, int out_size, void* d_ws, size_t ws_size,
                              hipStream_t stream)
{
    const float* scores = (const float*)d_in[0];   // [B,T,L] f32
    const int*   gold   = (const int*)  d_in[1];   // [B,T]
    const int*   mask   = (const int*)  d_in[2];   // [B,T]
    const float* trans  = (const float*)d_in[3];   // [L,L] f32
    float* partial = (float*)d_ws;                 // NG floats of scratch
    float* out     = (float*)d_out;

    crf_fwd_kernel<<<NG, 32, 0, stream>>>(scores, gold, mask, trans, partial);
    crf_final_kernel<<<1, 32, 0, stream>>>(partial, out);
}